// GNN_10222022164871
// MI455X (gfx1250) — compile-verified
//
#include <hip/hip_runtime.h>
#include <math.h>

typedef __attribute__((ext_vector_type(2))) float v2f;
typedef __attribute__((ext_vector_type(8))) float v8f;

#define NUM_GRAPHS 256
#define NEG_SLOPE 0.2f

// ---- monotonic float <-> uint encoding for atomicMax on floats ----
__device__ __forceinline__ unsigned f32_ord(float x) {
  unsigned u = __float_as_uint(x);
  return (u & 0x80000000u) ? ~u : (u | 0x80000000u);
}
__device__ __forceinline__ float ord_f32(unsigned u) {
  return (u & 0x80000000u) ? __uint_as_float(u & 0x7fffffffu) : __uint_as_float(~u);
}
// f32_ord(-inf) == ~0xFF800000 == 0x007FFFFF
#define ORD_NEG_INF 0x007FFFFFu

// =======================================================================
// GEMM: C[N,64] = A[N,64] @ W[64,64] (+ bias), via V_WMMA_F32_16X16X4_F32.
// Each wave32 computes one 16x16 tile; block = 8 waves = 32 rows x 64 cols.
// Fragment layouts per CDNA5 ISA 7.12.2 (fp32, 16x4 A / 4x16 B / 16x16 C).
// =======================================================================
__global__ void gemm_n64_wmma(const float* __restrict__ A,
                              const float* __restrict__ W,
                              const float* __restrict__ bias,
                              float* __restrict__ C, int N) {
  const int lane = threadIdx.x & 31;
  const int wave = threadIdx.x >> 5;
  const int row0 = blockIdx.x * 32 + (wave >> 2) * 16;
  const int col0 = (wave & 3) * 16;
  if (row0 >= N) return;  // wave-uniform: EXEC stays all-ones for WMMA

  const int mh  = lane & 15;        // row (A) / col (B) within tile
  const int khi = (lane >> 4) * 2;  // lanes 0-15 -> K{0,1}, lanes 16-31 -> K{2,3}

  int rA = row0 + mh;
  if (rA > N - 1) rA = N - 1;  // clamp (no divergence)
  const float* Arow = A + (size_t)rA * 64;

  v8f acc = {};
#pragma unroll
  for (int k0 = 0; k0 < 64; k0 += 4) {
    const int kk = k0 + khi;
    v2f a, b;
    a.x = Arow[kk];
    a.y = Arow[kk + 1];
    b.x = W[(size_t)kk * 64 + col0 + mh];
    b.y = W[(size_t)(kk + 1) * 64 + col0 + mh];
    acc = __builtin_amdgcn_wmma_f32_16x16x4_f32(false, a, false, b,
                                                (short)0, acc, false, false);
  }

  const int colg  = col0 + mh;
  const int rbase = (lane >> 4) * 8;  // lanes 0-15: M 0..7, lanes 16-31: M 8..15
  const float badd = bias ? bias[colg] : 0.0f;
#pragma unroll
  for (int r = 0; r < 8; ++r) {
    const int rowg = row0 + rbase + r;
    if (rowg < N) C[(size_t)rowg * 64 + colg] = acc[r] + badd;
  }
}

// =======================================================================
// Per-layer scatter/gather kernels (memory-bound part of the roofline)
// =======================================================================
__global__ void init_node_bufs(unsigned* __restrict__ maxb,
                               float* __restrict__ denom,
                               float* __restrict__ out, int N) {
  const int i = blockIdx.x * blockDim.x + threadIdx.x;
  const int nf = N * 64;
  if (i < nf) out[i] = 0.0f;
  if (i < N) { maxb[i] = ORD_NEG_INF; denom[i] = 0.0f; }
}

// wave32 per edge: e = att . leaky_relu(xl[src] + xr[dst]); atomicMax per dst
__global__ void edge_logits(const int* __restrict__ ei, int E, int Etot,
                            const float* __restrict__ xl,
                            const float* __restrict__ xr,
                            const float* __restrict__ att,
                            float* __restrict__ e_raw,
                            unsigned* __restrict__ maxb) {
  const int widx = (blockIdx.x * blockDim.x + threadIdx.x) >> 5;
  const int lane = threadIdx.x & 31;
  if (widx >= Etot) return;
  int src, dst;
  if (widx < E) { src = ei[widx]; dst = ei[E + widx]; }
  else          { src = dst = widx - E; }            // self-loop
  const float* pl = xl + (size_t)src * 64;
  const float* pr = xr + (size_t)dst * 64;
  float e = 0.0f;
#pragma unroll
  for (int t = 0; t < 2; ++t) {
    const int f = lane + t * 32;
    float m = pl[f] + pr[f];
    m = (m > 0.0f) ? m : NEG_SLOPE * m;
    e += m * att[f];
  }
#pragma unroll
  for (int off = 16; off > 0; off >>= 1) e += __shfl_xor(e, off, 32);
  if (lane == 0) {
    e_raw[widx] = e;
    atomicMax(maxb + dst, f32_ord(e));
  }
}

// thread per edge: e_exp = exp(e - max[dst]); atomicAdd into denom[dst]
__global__ void edge_exp(const int* __restrict__ ei, int E, int Etot,
                         const float* __restrict__ e_raw,
                         const unsigned* __restrict__ maxb,
                         float* __restrict__ e_exp,
                         float* __restrict__ denom) {
  const int i = blockIdx.x * blockDim.x + threadIdx.x;
  if (i >= Etot) return;
  const int dst = (i < E) ? ei[E + i] : (i - E);
  const float ee = __expf(e_raw[i] - ord_f32(maxb[dst]));
  e_exp[i] = ee;
  atomicAdd(denom + dst, ee);
}

// wave32 per edge: out[dst] += alpha * xl[src]
__global__ void edge_scatter(const int* __restrict__ ei, int E, int Etot,
                             const float* __restrict__ xl,
                             const float* __restrict__ e_exp,
                             const float* __restrict__ denom,
                             float* __restrict__ out) {
  const int widx = (blockIdx.x * blockDim.x + threadIdx.x) >> 5;
  const int lane = threadIdx.x & 31;
  if (widx >= Etot) return;
  int src, dst;
  if (widx < E) { src = ei[widx]; dst = ei[E + widx]; }
  else          { src = dst = widx - E; }
  const float alpha = e_exp[widx] / denom[dst];
  const float* pl = xl + (size_t)src * 64;
  float* po = out + (size_t)dst * 64;
  atomicAdd(po + lane,      alpha * pl[lane]);
  atomicAdd(po + lane + 32, alpha * pl[lane + 32]);
}

__global__ void bias_act(float* __restrict__ h, const float* __restrict__ bias,
                         int N, int relu) {
  const int i = blockIdx.x * blockDim.x + threadIdx.x;
  if (i >= N * 64) return;
  float v = h[i] + bias[i & 63];
  if (relu) v = fmaxf(v, 0.0f);
  h[i] = v;
}

// =======================================================================
// Global mean pooling over graphs
// =======================================================================
__global__ void pool_zero(float* __restrict__ pooled, float* __restrict__ cnts) {
  const int i = blockIdx.x * blockDim.x + threadIdx.x;
  if (i < NUM_GRAPHS * 64) pooled[i] = 0.0f;
  if (i < NUM_GRAPHS) cnts[i] = 0.0f;
}

__global__ void pool_accum(const float* __restrict__ h,
                           const int* __restrict__ batch,
                           float* __restrict__ pooled,
                           float* __restrict__ cnts, int N) {
  const int i = blockIdx.x * blockDim.x + threadIdx.x;
  if (i >= N * 64) return;
  const int node = i >> 6, f = i & 63;
  const int g = batch[node];
  atomicAdd(pooled + g * 64 + f, h[i]);
  if (f == 0) atomicAdd(cnts + g, 1.0f);
}

__global__ void pool_div(float* __restrict__ pooled, const float* __restrict__ cnts) {
  const int i = blockIdx.x * blockDim.x + threadIdx.x;
  if (i >= NUM_GRAPHS * 64) return;
  pooled[i] /= fmaxf(cnts[i >> 6], 1.0f);
}

// =======================================================================
// Host-side layer driver + entry point
// =======================================================================
static void run_gat_layer(const int* ei, int E, int Etot, int N,
                          const float* xl, const float* xr, const float* att,
                          const float* bias, int relu, float* out,
                          float* e_raw, float* e_exp, unsigned* maxb,
                          float* denom, hipStream_t stream) {
  const int gridNF = (N * 64 + 255) / 256;
  const int gridE  = (Etot + 255) / 256;
  const int gridEW = (int)(((long long)Etot * 32 + 255) / 256);  // wave/edge
  init_node_bufs<<<gridNF, 256, 0, stream>>>(maxb, denom, out, N);
  edge_logits<<<gridEW, 256, 0, stream>>>(ei, E, Etot, xl, xr, att, e_raw, maxb);
  edge_exp<<<gridE, 256, 0, stream>>>(ei, E, Etot, e_raw, maxb, e_exp, denom);
  edge_scatter<<<gridEW, 256, 0, stream>>>(ei, E, Etot, xl, e_exp, denom, out);
  bias_act<<<gridNF, 256, 0, stream>>>(out, bias, N, relu);
}

extern "C" void kernel_launch(void* const* d_in, const int* in_sizes, int n_in,
                              void* d_out, int out_size, void* d_ws, size_t ws_size,
                              hipStream_t stream) {
  const float* x    = (const float*)d_in[0];
  const int*   ei   = (const int*)d_in[1];
  const int*   bat  = (const int*)d_in[2];
  const float* W1l  = (const float*)d_in[3];
  const float* W1r  = (const float*)d_in[4];
  const float* att1 = (const float*)d_in[5];
  const float* b1   = (const float*)d_in[6];
  const float* W2   = (const float*)d_in[7];
  const float* att2 = (const float*)d_in[8];
  const float* b2   = (const float*)d_in[9];
  const float* W3   = (const float*)d_in[10];
  const float* att3 = (const float*)d_in[11];
  const float* b3   = (const float*)d_in[12];
  const float* Wlin = (const float*)d_in[13];
  const float* blin = (const float*)d_in[14];

  const int N    = in_sizes[0] / 64;
  const int E    = in_sizes[1] / 2;
  const int Etot = E + N;  // + self loops

  // workspace carve-up (~91 MB for N=100k, E=1.6M)
  float* bufA   = (float*)d_ws;                 // N*64  (xl)
  float* bufB   = bufA + (size_t)N * 64;        // N*64  (xr / layer2 out)
  float* bufC   = bufB + (size_t)N * 64;        // N*64  (layer1/3 out)
  float* e_raw  = bufC + (size_t)N * 64;        // Etot
  float* e_exp  = e_raw + Etot;                 // Etot
  unsigned* maxb = (unsigned*)(e_exp + Etot);   // N
  float* denom  = (float*)(maxb + N);           // N
  float* pooled = denom + N;                    // 256*64
  float* cnts   = pooled + NUM_GRAPHS * 64;     // 256

  const int gemmGrid = (N + 31) / 32;
  const int gridNF   = (N * 64 + 255) / 256;

  // Layer 1: separate Wl / Wr
  gemm_n64_wmma<<<gemmGrid, 256, 0, stream>>>(x, W1l, nullptr, bufA, N);
  gemm_n64_wmma<<<gemmGrid, 256, 0, stream>>>(x, W1r, nullptr, bufB, N);
  run_gat_layer(ei, E, Etot, N, bufA, bufB, att1, b1, 1, bufC,
                e_raw, e_exp, maxb, denom, stream);

  // Layer 2: shared weights -> one GEMM, xl == xr
  gemm_n64_wmma<<<gemmGrid, 256, 0, stream>>>(bufC, W2, nullptr, bufA, N);
  run_gat_layer(ei, E, Etot, N, bufA, bufA, att2, b2, 1, bufB,
                e_raw, e_exp, maxb, denom, stream);

  // Layer 3: shared weights, no relu
  gemm_n64_wmma<<<gemmGrid, 256, 0, stream>>>(bufB, W3, nullptr, bufA, N);
  run_gat_layer(ei, E, Etot, N, bufA, bufA, att3, b3, 0, bufC,
                e_raw, e_exp, maxb, denom, stream);

  // Global mean pool + head GEMM (WMMA, fused bias) straight into d_out
  pool_zero<<<(NUM_GRAPHS * 64 + 255) / 256, 256, 0, stream>>>(pooled, cnts);
  pool_accum<<<gridNF, 256, 0, stream>>>(bufC, bat, pooled, cnts, N);
  pool_div<<<(NUM_GRAPHS * 64 + 255) / 256, 256, 0, stream>>>(pooled, cnts);
  gemm_n64_wmma<<<(NUM_GRAPHS + 31) / 32, 256, 0, stream>>>(pooled, Wlin, blin,
                                                            (float*)d_out, NUM_GRAPHS);
}